// SageEncoder_47485158425283
// MI455X (gfx1250) — compile-verified
//
#include <hip/hip_runtime.h>

typedef __attribute__((ext_vector_type(2))) float v2f;
typedef __attribute__((ext_vector_type(8))) float v8f;

#define N_NODES 100000
#define D 64
#define EPS 1e-5f

// ---------------------------------------------------------------------------
// Scatter: agg[dst] += h[src], cnt[dst] += 1  (mean-aggregation numerator/denominator)
// One thread per (edge, 4-feature chunk).
// ---------------------------------------------------------------------------
__global__ __launch_bounds__(256)
void sage_scatter(const float* __restrict__ h,
                  const int* __restrict__ src,
                  const int* __restrict__ dst,
                  float* __restrict__ agg,
                  float* __restrict__ cnt,
                  int nEdges) {
    long long tid = (long long)blockIdx.x * blockDim.x + threadIdx.x;
    long long total = (long long)nEdges * 16;
    if (tid >= total) return;
    int e = (int)(tid >> 4);
    int g = (int)(tid & 15);
    int s = src[e];
    int d = dst[e];
    const float4 v = *(const float4*)(h + (long long)s * D + g * 4);
    float* a = agg + (long long)d * D + g * 4;
    atomicAdd(a + 0, v.x);
    atomicAdd(a + 1, v.y);
    atomicAdd(a + 2, v.z);
    atomicAdd(a + 3, v.w);
    if (g == 0) atomicAdd(cnt + d, 1.0f);
}

// ---------------------------------------------------------------------------
// WMMA GEMM: out = mean @ Wl^T + bl + h @ Wr^T   (fp32 V_WMMA_F32_16X16X4_F32)
// One wave per 16x16 output tile; 25000 tiles total; 8 waves / block.
// RELU_STATS: apply ReLU and accumulate per-feature sum / sumsq for GraphNorm.
// ---------------------------------------------------------------------------
template <bool RELU_STATS>
__global__ __launch_bounds__(256)
void sage_gemm(const float* __restrict__ agg, const float* __restrict__ cnt,
               const float* __restrict__ h,
               const float* __restrict__ Wl, const float* __restrict__ bl,
               const float* __restrict__ Wr,
               float* __restrict__ out, float* __restrict__ stats) {
    __shared__ float s_sum[D];
    __shared__ float s_sq[D];
    if (RELU_STATS) {
        if (threadIdx.x < D) { s_sum[threadIdx.x] = 0.0f; s_sq[threadIdx.x] = 0.0f; }
        __syncthreads();
    }

    const int lane = threadIdx.x & 31;
    const int wave = threadIdx.x >> 5;
    const int tile = blockIdx.x * 8 + wave;      // [0, 25000)
    const int mt   = tile >> 2;                  // 16-row tile of nodes
    const int nb   = (tile & 3) * 16;            // 16-col tile of features
    const int mrow = mt * 16 + (lane & 15);      // node row this lane feeds A from
    const int koff = (lane >> 4) * 2;            // lanes 16..31 carry K+2,K+3
    const int ncol = nb + (lane & 15);           // output feature column

    const float rc = 1.0f / fmaxf(cnt[mrow], 1.0f);

    // accumulator initialized with bias (broadcast per output column)
    const float bias = bl[ncol];
    v8f acc;
    for (int r = 0; r < 8; ++r) acc[r] = bias;

    const float* aggRow = agg + (long long)mrow * D;
    const float* hRow   = h   + (long long)mrow * D;

    // mean @ Wl^T : A[m][k] = agg[m][k]*rc, B[k][n] = Wl[n][k]
    for (int k0 = 0; k0 < D; k0 += 4) {
        const int kk = k0 + koff;
        v2f a; a.x = aggRow[kk] * rc; a.y = aggRow[kk + 1] * rc;
        const float* wrow = Wl + ncol * D + kk;
        v2f b; b.x = wrow[0]; b.y = wrow[1];
        acc = __builtin_amdgcn_wmma_f32_16x16x4_f32(false, a, false, b,
                                                    (short)0, acc, false, false);
    }
    // h @ Wr^T
    for (int k0 = 0; k0 < D; k0 += 4) {
        const int kk = k0 + koff;
        v2f a; a.x = hRow[kk]; a.y = hRow[kk + 1];
        const float* wrow = Wr + ncol * D + kk;
        v2f b; b.x = wrow[0]; b.y = wrow[1];
        acc = __builtin_amdgcn_wmma_f32_16x16x4_f32(false, a, false, b,
                                                    (short)0, acc, false, false);
    }

    // C/D layout: VGPR r -> row r (lanes 0-15) or r+8 (lanes 16-31), col = lane&15
    const int rbase = mt * 16 + ((lane >> 4) * 8);
    for (int r = 0; r < 8; ++r) {
        float v = acc[r];
        if (RELU_STATS) {
            v = fmaxf(v, 0.0f);
            atomicAdd(&s_sum[ncol - nb + nb], v);   // feature index = ncol (0..63)
            atomicAdd(&s_sq[ncol], v * v);
        }
        out[(long long)(rbase + r) * D + ncol] = v;
    }

    if (RELU_STATS) {
        __syncthreads();
        if (threadIdx.x < D)
            atomicAdd(&stats[threadIdx.x], s_sum[threadIdx.x]);
        else if (threadIdx.x < 2 * D)
            atomicAdd(&stats[threadIdx.x], s_sq[threadIdx.x - D]);
    }
}

// ---------------------------------------------------------------------------
// GraphNorm coefficients: fold weight/bias/scale + mean/var into A*h + B
// ---------------------------------------------------------------------------
__global__ void gn_coef(float* __restrict__ stats,
                        const float* __restrict__ w,
                        const float* __restrict__ b,
                        const float* __restrict__ scale) {
    const int d = threadIdx.x;
    const float invN = 1.0f / (float)N_NODES;
    const float mu = stats[d] * invN;
    const float m2 = stats[D + d] * invN;
    const float s  = scale[d];
    // var of (h - s*mu) over nodes = E[h^2] - s*mu^2*(2 - s)
    const float var = m2 - s * mu * mu * (2.0f - s);
    const float inv = rsqrtf(var + EPS);
    const float A = w[d] * inv;
    const float B = b[d] - A * s * mu;
    stats[d]     = A;
    stats[D + d] = B;
}

__global__ __launch_bounds__(256)
void gn_apply(float* __restrict__ h, const float* __restrict__ coef) {
    long long i = (long long)blockIdx.x * blockDim.x + threadIdx.x;
    if (i >= (long long)N_NODES * D) return;
    int d = (int)(i & (D - 1));
    h[i] = coef[d] * h[i] + coef[D + d];
}

// ---------------------------------------------------------------------------
extern "C" void kernel_launch(void* const* d_in, const int* in_sizes, int n_in,
                              void* d_out, int out_size, void* d_ws, size_t ws_size,
                              hipStream_t stream) {
    const float* x  = (const float*)d_in[0];
    const int*   ei = (const int*)d_in[1];     // [2, E]
    const float* Wl = (const float*)d_in[2];   // [3, 64, 64]
    const float* bl = (const float*)d_in[3];   // [3, 64]
    const float* Wr = (const float*)d_in[4];   // [3, 64, 64]
    const float* gw = (const float*)d_in[5];
    const float* gb = (const float*)d_in[6];
    const float* gs = (const float*)d_in[7];

    const int E = in_sizes[1] / 2;
    const int* src = ei;
    const int* dst = ei + E;

    float* ws    = (float*)d_ws;
    float* agg   = ws;                                 // N*D
    float* cnt   = agg + (size_t)N_NODES * D;          // N
    float* hA    = cnt + N_NODES;                      // N*D
    float* hB    = hA + (size_t)N_NODES * D;           // N*D
    float* stats = hB + (size_t)N_NODES * D;           // 2*D

    float* out = (float*)d_out;

    const float* hin[3]  = {x, hA, hB};
    float*       hout[3] = {hA, hB, out};

    const int scatterBlocks = (int)(((long long)E * 16 + 255) / 256);
    const int gemmBlocks    = (N_NODES / 16 * 4) / 8;      // 3125
    const int normBlocks    = (N_NODES * D) / 256;         // 25000

    for (int l = 0; l < 3; ++l) {
        hipMemsetAsync(agg, 0, (size_t)N_NODES * D * sizeof(float), stream);
        hipMemsetAsync(cnt, 0, (size_t)N_NODES * sizeof(float), stream);
        sage_scatter<<<scatterBlocks, 256, 0, stream>>>(hin[l], src, dst, agg, cnt, E);
        if (l < 2) {
            hipMemsetAsync(stats, 0, 2 * D * sizeof(float), stream);
            sage_gemm<true><<<gemmBlocks, 256, 0, stream>>>(
                agg, cnt, hin[l], Wl + l * D * D, bl + l * D, Wr + l * D * D,
                hout[l], stats);
            gn_coef<<<1, D, 0, stream>>>(stats, gw, gb, gs);
            gn_apply<<<normBlocks, 256, 0, stream>>>(hout[l], stats);
        } else {
            sage_gemm<false><<<gemmBlocks, 256, 0, stream>>>(
                agg, cnt, hin[l], Wl + l * D * D, bl + l * D, Wr + l * D * D,
                hout[l], nullptr);
        }
    }
}